// TokenClassificationCrfHead_66383014527259
// MI455X (gfx1250) — compile-verified
//
#include <hip/hip_runtime.h>
#include <hip/hip_bf16.h>
#include <math.h>

// Problem constants (from reference): B=64, S=512, H=1024, L=9
#define B_  64
#define S_  512
#define H_  1024
#define L_  9
#define M_  (B_ * S_)          // 32768 rows of the emissions GEMM

typedef __attribute__((ext_vector_type(2))) float v2f;
typedef __attribute__((ext_vector_type(8))) float v8f;

// One pipeline "body" = K-chunk of 32 (8 WMMA steps of K=4 each).
struct Body {
    v2f a[8];
    v2f b[8];
};

// Issue all 16 b64 loads of one body (forms a load clause; 16 outstanding).
__device__ __forceinline__ void load_body(Body& f,
                                          const float* __restrict__ arow,
                                          const float* __restrict__ brow,
                                          int ko) {
    #pragma unroll
    for (int j = 0; j < 8; ++j) {
        f.a[j] = *(const v2f*)(arow + ko + 4 * j);
        f.b[j] = *(const v2f*)(brow + ko + 4 * j);
    }
}

// Consume one body: 8 x v_wmma_f32_16x16x4_f32 (exact f32, K=4 per issue).
__device__ __forceinline__ void mac_body(v8f& acc, const Body& f) {
    #pragma unroll
    for (int j = 0; j < 8; ++j)
        acc = __builtin_amdgcn_wmma_f32_16x16x4_f32(false, f.a[j], false, f.b[j],
                                                    (short)0, acc, false, false);
}

// ---------------------------------------------------------------------------
// Kernel 1: emissions[m][l] = dot(X[m,:], W[l,:]) + bias[l]
//   A (16x4 f32): lane l -> row M = l&15; VGPR0/1 = K = (l>>4)*2 + {0,1}
//   B (4x16 f32): lane l -> col N = l&15; same K striping.
// One wave per 16x16 tile -> 2048 waves (max latency hiding for the
// HBM-bound 128 MB stream; floor ~5.8 us at 23.3 TB/s).
// Double-buffered bodies keep 16 loads in flight across each WMMA block
// (s_wait_loadcnt ~16, never 0). Columns 9..15 computed on clamped W row 0
// and discarded at store, so the K-loop is branch-free (EXEC all-1s).
// ---------------------------------------------------------------------------
__global__ void emissions_gemm_wmma(const float* __restrict__ X,
                                    const float* __restrict__ W,
                                    const float* __restrict__ bias,
                                    float* __restrict__ em) {
    const int lane = threadIdx.x & 31;
    const int wave = threadIdx.x >> 5;
    const int task = blockIdx.x * 8 + wave;           // 2048 wave-tasks
    const int row0 = task * 16;                       // 16-row tile base
    const int n    = lane & 15;                       // A: row-in-tile; B/C/D: column
    const int hi   = lane >> 4;                       // lane group
    const int kofs = hi * 2;                          // K offset within a 4-step

    const float* __restrict__ arow = X + (size_t)(row0 + n) * H_;
    const float* __restrict__ brow = W + (size_t)(n < L_ ? n : 0) * H_;

    Body f0, f1;
    load_body(f0, arow, brow, 0 + kofs);
    load_body(f1, arow, brow, 32 + kofs);

    v8f acc = {};
    for (int kc = 0; kc < H_ - 64; kc += 64) {
        __builtin_prefetch(arow + kc + 512, 0, 3);    // global_prefetch_b8, 2KB ahead
        mac_body(acc, f0);                            // consume kc..kc+31
        load_body(f0, arow, brow, kc + 64 + kofs);    // refill for kc+64
        mac_body(acc, f1);                            // consume kc+32..kc+63
        load_body(f1, arow, brow, kc + 96 + kofs);    // refill for kc+96
    }
    mac_body(acc, f0);                                // kc = H_-64
    mac_body(acc, f1);                                // kc = H_-32

    // C/D layout: lane l, VGPR r -> M = r + (l>>4)*8, N = l&15
    if (n < L_) {
        const float bv = bias[n];
        #pragma unroll
        for (int r = 0; r < 8; ++r)
            em[(size_t)(row0 + hi * 8 + r) * L_ + n] = acc[r] + bv;
    }
}

// ---------------------------------------------------------------------------
// Kernel 2: CRF NLL per batch element. One wave per sequence; lanes 0..8
// carry the L=9 alpha states, __shfl broadcasts alpha for the 9x9
// logsumexp. Serial over S=512 -> the per-step dependence chain is the
// wall clock: terms computed once, pairwise max/sum trees, emissions and
// mask prefetched 16 and 64 steps ahead (resp.). partial[b] = gold_score(b) - logZ(b).
// ---------------------------------------------------------------------------
__global__ void crf_nll_kernel(const float* __restrict__ em,
                               const int* __restrict__ tags,
                               const unsigned char* __restrict__ mask,
                               const float* __restrict__ startT,
                               const float* __restrict__ endT,
                               const float* __restrict__ trans,
                               float* __restrict__ partial) {
    const int b    = blockIdx.x;
    const int lane = threadIdx.x;            // 0..31
    const bool act = (lane < L_);

    const float* __restrict__ emb = em + (size_t)b * S_ * L_;
    const int* __restrict__ tg   = tags + b * S_;
    const unsigned char* __restrict__ mk = mask + b * S_;

    const float NEG = -3.0e38f;

    // Column lane of the transition matrix: Tcol[i] = trans[i][lane]
    float Tcol[L_];
    #pragma unroll
    for (int i = 0; i < L_; ++i) Tcol[i] = act ? trans[i * L_ + lane] : 0.f;

    // alpha0 = start + emissions[0]
    float alpha = act ? (startT[lane] + emb[lane]) : NEG;

    for (int t = 1; t < S_; ++t) {
        __builtin_prefetch(emb + (t + 16) * L_ + lane, 0, 3);   // stay in WGP$
        __builtin_prefetch(mk + t + 64, 0, 3);

        float v[L_];
        #pragma unroll
        for (int i = 0; i < L_; ++i) v[i] = __shfl(alpha, i, 32) + Tcol[i];

        // pairwise max tree (depth 4 instead of serial 9)
        const float m01 = fmaxf(v[0], v[1]), m23 = fmaxf(v[2], v[3]);
        const float m45 = fmaxf(v[4], v[5]), m67 = fmaxf(v[6], v[7]);
        const float mA  = fmaxf(m01, m23),   mB  = fmaxf(m45, m67);
        const float mx  = fmaxf(fmaxf(mA, mB), v[8]);

        float e[L_];
        #pragma unroll
        for (int i = 0; i < L_; ++i) e[i] = __expf(v[i] - mx);
        // pairwise sum tree
        const float s01 = e[0] + e[1], s23 = e[2] + e[3];
        const float s45 = e[4] + e[5], s67 = e[6] + e[7];
        const float s   = ((s01 + s23) + (s45 + s67)) + e[8];

        const float ev  = act ? emb[t * L_ + lane] : 0.f;
        const float nxt = mx + __logf(s) + ev;
        if (mk[t] && act) alpha = nxt;       // masked update, uniform per (b,t)
    }

    // logZ = logsumexp(alpha + end) over the 9 states (inactive lanes = -inf)
    float vz = act ? (alpha + endT[lane]) : NEG;
    float mx = vz;
    #pragma unroll
    for (int off = 16; off >= 1; off >>= 1) mx = fmaxf(mx, __shfl_xor(mx, off, 32));
    float s = act ? __expf(vz - mx) : 0.f;
    #pragma unroll
    for (int off = 16; off >= 1; off >>= 1) s += __shfl_xor(s, off, 32);
    const float logZ = mx + __logf(s);

    // Gold path score, timesteps strided across lanes
    float sc  = 0.f;
    int   cnt = 0;
    for (int t = lane; t < S_; t += 32) {
        cnt += mk[t] ? 1 : 0;
        if (t == 0) {
            sc += startT[tg[0]] + emb[tg[0]];
        } else if (mk[t]) {
            sc += trans[tg[t - 1] * L_ + tg[t]] + emb[t * L_ + tg[t]];
        }
    }
    #pragma unroll
    for (int off = 16; off >= 1; off >>= 1) {
        sc  += __shfl_xor(sc, off, 32);
        cnt += __shfl_xor(cnt, off, 32);
    }

    if (lane == 0) {
        const int last = tg[cnt - 1];        // seq_end = sum(mask) - 1
        partial[b] = (sc + endT[last]) - logZ;
    }
}

// ---------------------------------------------------------------------------
// Kernel 3: deterministic final reduction: out = -(sum_b partial[b])
// ---------------------------------------------------------------------------
__global__ void crf_finalize(const float* __restrict__ partial, float* __restrict__ out) {
    const int lane = threadIdx.x;            // 0..31, B_ = 64 = 2 * 32
    float v = partial[lane] + partial[lane + 32];
    #pragma unroll
    for (int off = 16; off >= 1; off >>= 1) v += __shfl_xor(v, off, 32);
    if (lane == 0) out[0] = -v;
}

extern "C" void kernel_launch(void* const* d_in, const int* in_sizes, int n_in,
                              void* d_out, int out_size, void* d_ws, size_t ws_size,
                              hipStream_t stream) {
    const float*         X      = (const float*)d_in[0];          // (B,S,H) f32
    const int*           tags   = (const int*)d_in[1];            // (B,S) i32
    const unsigned char* mask   = (const unsigned char*)d_in[2];  // (B,S) bool
    const float*         W      = (const float*)d_in[3];          // (L,H) f32
    const float*         bias   = (const float*)d_in[4];          // (L,)
    const float*         startT = (const float*)d_in[5];          // (L,)
    const float*         endT   = (const float*)d_in[6];          // (L,)
    const float*         trans  = (const float*)d_in[7];          // (L,L)

    float* em      = (float*)d_ws;                 // B*S*L floats (~1.13 MB), fully written below
    float* partial = em + (size_t)M_ * L_;         // B floats
    float* out     = (float*)d_out;                // scalar NLL

    // 256 blocks x 256 threads: 8 waves/block x 16 rows/wave -> 2048 waves
    emissions_gemm_wmma<<<M_ / (16 * 8), 256, 0, stream>>>(X, W, bias, em);
    crf_nll_kernel<<<B_, 32, 0, stream>>>(em, tags, mask, startT, endT, trans, partial);
    crf_finalize<<<1, 32, 0, stream>>>(partial, out);
}